// RGSL_64561948393776
// MI455X (gfx1250) — compile-verified
//
#include <hip/hip_runtime.h>
#include <hip/hip_bf16.h>
#include <math.h>

// ---------------- problem constants ----------------
#define NB 32      // batch
#define TT 12      // timesteps
#define NN 512     // nodes
#define DIN 1
#define HH 64
#define KK 3
#define ED 10
#define HOR 12

// padded per-k channel widths (multiple of 16)
#define CP0 80     // layer0: DIN+H = 65 -> 80
#define CP1 128    // layer1: 2H = 128

#define KC 64      // K-chunk staged in LDS for the 512-deep GEMMs

typedef __attribute__((ext_vector_type(2))) float v2f;
typedef __attribute__((ext_vector_type(8))) float v8f;

// =====================================================================
// 1) adjacency from gumbel-softmax hard sample + row degree
// =====================================================================
__global__ __launch_bounds__(256) void adj_kernel(const float* __restrict__ emb,
                                                  const float* __restrict__ gn,
                                                  float* __restrict__ adj,
                                                  float* __restrict__ dd) {
    __shared__ float sred[256];
    int n = blockIdx.x;
    float en[ED]; float nn = 0.f;
#pragma unroll
    for (int q = 0; q < ED; q++) { en[q] = emb[n * ED + q]; nn += en[q] * en[q]; }
    nn = sqrtf(nn);
    float sum = 0.f;
    for (int m = threadIdx.x; m < NN; m += 256) {
        float dot = 0.f, nm = 0.f;
#pragma unroll
        for (int q = 0; q < ED; q++) { float em = emb[m * ED + q]; dot += en[q] * em; nm += em * em; }
        nm = sqrtf(nm);
        float lg = (dot / (nn * nm) + 1.f) * 0.5f;
        long gi = ((long)n * NN + m) * 2;
        float y0 = lg + gn[gi + 0];
        float y1 = (1.f - lg) + gn[gi + 1];
        float a = (m == n) ? 0.f : ((y0 >= y1) ? 1.f : 0.f);  // argmax ties -> idx 0
        adj[(long)n * NN + m] = a;
        sum += a;
    }
    sred[threadIdx.x] = sum; __syncthreads();
    for (int off = 128; off > 0; off >>= 1) {
        if (threadIdx.x < off) sred[threadIdx.x] += sred[threadIdx.x + off];
        __syncthreads();
    }
    if (threadIdx.x == 0) dd[n] = sred[0];
}

// =====================================================================
// 2) lam = max(L) - min(L); L = -adj + diag(d)
// =====================================================================
__global__ __launch_bounds__(512) void lam_kernel(const float* __restrict__ dd,
                                                  float* __restrict__ lam) {
    __shared__ float smx[512], smn[512];
    int i = threadIdx.x;
    float v = dd[i];
    smx[i] = v; smn[i] = v; __syncthreads();
    for (int off = 256; off > 0; off >>= 1) {
        if (i < off) { smx[i] = fmaxf(smx[i], smx[i + off]); smn[i] = fminf(smn[i], smn[i + off]); }
        __syncthreads();
    }
    if (i == 0) {
        float maxL = smx[0];
        float minL = fminf(smn[0], (maxL > 0.f) ? -1.f : 0.f);
        lam[0] = maxL - minL;
    }
}

// =====================================================================
// 3) tilde = 2*L/lam - I
// =====================================================================
__global__ __launch_bounds__(256) void tilde_kernel(const float* __restrict__ adj,
                                                    const float* __restrict__ dd,
                                                    const float* __restrict__ lam,
                                                    float* __restrict__ tld) {
    long idx = (long)blockIdx.x * 256 + threadIdx.x;
    int n = (int)(idx >> 9), m = (int)(idx & (NN - 1));
    float L = (n == m) ? dd[n] : -adj[idx];
    tld[idx] = 2.f * L / lam[0] - ((n == m) ? 1.f : 0.f);
}

// =====================================================================
// 4) LDS-staged 512-deep WMMA GEMM.
//    C[m, c] = sum_k A[m,k] * B[k,c]
//    A: (512,512) row-major (a graph support).
//    B: per-z-slice base Bbase + z*Bbstr, row stride ldb.
//    O: per-z-slice base Obase + z*Obstr, row stride ldo.
//    post: 0 -> store acc ; 1 -> store 2*acc - I (Chebyshev recurrence)
//    grid (4, ncols/16, nz), block (32,8): block computes 128 x 16 slab.
// =====================================================================
__global__ __launch_bounds__(256) void gemm512_kernel(const float* __restrict__ A,
                                                      const float* __restrict__ Bbase, long Bbstr, int ldb,
                                                      float* __restrict__ Obase, long Obstr, int ldo,
                                                      int post) {
    __shared__ float sA[128][KC + 1];   // +1 pad: 65 = 1 mod 64 banks -> conflict-free
    __shared__ float sB[16][KC + 1];
    int lane = threadIdx.x, wave = threadIdx.y;
    int tid = wave * 32 + lane;
    int r = lane & 15, half = lane >> 4;
    int rowBase = blockIdx.x * 128;
    int tc = blockIdx.y * 16;
    long z = blockIdx.z;
    const float* Bm = Bbase + z * Bbstr;
    float* Om = Obase + z * Obstr;

    v8f acc = {};
    for (int k0 = 0; k0 < NN; k0 += KC) {
        // ---- stage A panel: 128 x KC (float4 loads, 8 per thread) ----
#pragma unroll
        for (int i = 0; i < 8; i++) {
            int e4 = (i * 256 + tid);           // float4 index within panel
            int row = e4 >> 4;                  // KC/4 = 16 float4 per row
            int kk = (e4 & 15) * 4;
            float4 v = *(const float4*)(&A[(long)(rowBase + row) * NN + k0 + kk]);
            sA[row][kk + 0] = v.x; sA[row][kk + 1] = v.y;
            sA[row][kk + 2] = v.z; sA[row][kk + 3] = v.w;
        }
        // ---- stage B panel: KC x 16, transposed into sB[c][k] ----
#pragma unroll
        for (int i = 0; i < 4; i++) {
            int j = i * 256 + tid;
            int kk = j >> 4;                    // 0..63
            int c = j & 15;
            sB[c][kk] = Bm[(long)(k0 + kk) * ldb + tc + c];
        }
        // ---- prefetch next chunk while we compute (global_prefetch_b8) ----
        if (k0 + KC < NN) {
            __builtin_prefetch(&A[(long)(rowBase + (tid & 127)) * NN + k0 + KC], 0, 1);
            __builtin_prefetch(&Bm[(long)(k0 + KC + (tid & 63)) * ldb + tc], 0, 1);
        }
        __syncthreads();
        // ---- 16 WMMA steps off LDS ----
        int arow = wave * 16 + r;
#pragma unroll
        for (int kk = 0; kk < KC; kk += 4) {
            v2f a, bb;
            a.x = sA[arow][kk + 2 * half];
            a.y = sA[arow][kk + 2 * half + 1];
            bb.x = sB[r][kk + 2 * half];
            bb.y = sB[r][kk + 2 * half + 1];
            acc = __builtin_amdgcn_wmma_f32_16x16x4_f32(false, a, false, bb, (short)0, acc, false, false);
        }
        __syncthreads();
    }
    int tm = rowBase + wave * 16;
#pragma unroll
    for (int v = 0; v < 8; v++) {
        int row = tm + v + 8 * half, col = tc + r;
        float val = acc[v];
        if (post == 1) val = 2.f * val - ((row == col) ? 1.f : 0.f);
        Om[(long)row * ldo + col] = val;
    }
}

// =====================================================================
// 5) per-node weights W[n,k,i,o] = sum_d emb[n,d]*wp[d,k,i,o] (i padded)
// =====================================================================
__global__ __launch_bounds__(256) void build_w_kernel(const float* __restrict__ emb,
                                                      const float* __restrict__ wp,
                                                      float* __restrict__ W,
                                                      int I, int Ip, int O) {
    int n = blockIdx.x;
    float e[ED];
#pragma unroll
    for (int q = 0; q < ED; q++) e[q] = emb[n * ED + q];
    int tot = KK * Ip * O;
    for (int idx = threadIdx.x; idx < tot; idx += 256) {
        int o = idx % O, ki = idx / O, k = ki / Ip, i = ki % Ip;
        float v = 0.f;
        if (i < I) {
#pragma unroll
            for (int q = 0; q < ED; q++)
                v += e[q] * wp[(((long)q * KK + k) * I + i) * O + o];
        }
        W[(long)n * tot + idx] = v;
    }
}

__global__ __launch_bounds__(256) void build_b_kernel(const float* __restrict__ emb,
                                                      const float* __restrict__ bp,
                                                      float* __restrict__ Bb, int O) {
    int n = blockIdx.x;
    for (int o = threadIdx.x; o < O; o += 256) {
        float v = 0.f;
#pragma unroll
        for (int q = 0; q < ED; q++) v += emb[n * ED + q] * bp[q * O + o];
        Bb[n * O + o] = v;
    }
}

// =====================================================================
// 6) concat -> xg k=0 slice: [x_t | (z?)*h | zero-pad]
// =====================================================================
__global__ __launch_bounds__(128) void concat_kernel(const float* __restrict__ x, long xbstr, int xdim,
                                                     const float* __restrict__ h,
                                                     const float* __restrict__ zr, // nullable; use cols [0,H)
                                                     float* __restrict__ xg, int Cp) {
    int n = blockIdx.x; long b = blockIdx.y;
    int n3 = 3 * Cp;
    float* dst = xg + b * (long)NN * n3 + (long)n * n3;
    for (int c = threadIdx.x; c < Cp; c += 128) {
        float v = 0.f;
        if (c < xdim) v = x[b * xbstr + (long)n * xdim + c];
        else if (c < xdim + HH) {
            float hv = h[b * (long)NN * HH + (long)n * HH + (c - xdim)];
            if (zr) hv *= zr[b * (long)NN * (2 * HH) + (long)n * (2 * HH) + (c - xdim)];
            v = hv;
        }
        dst[c] = v;
    }
}

// =====================================================================
// 7) per-node GEMM: out[b,n,o] = act( sum_ki xg[b,n,ki]*W[n,ki,o] + bias[n,o] )
//    A: M=32 (batch), K=3*Cp, B: W_n (3*Cp x O). grid = NN, block (32,8)
//    act: 0 none, 1 sigmoid, 2 tanh. out row stride per n = outN.
// =====================================================================
__global__ __launch_bounds__(256) void node_gemm_kernel(const float* __restrict__ xg,
                                                        const float* __restrict__ W,
                                                        const float* __restrict__ bias,
                                                        float* __restrict__ out,
                                                        int KI, int Ocols, int act, int outN) {
    int n = blockIdx.x;
    int lane = threadIdx.x, wave = threadIdx.y;
    int r = lane & 15, half = lane >> 4;
    const float* A = xg + (long)n * KI;          // A[b][ki] at b*NN*KI + n*KI + ki
    long Astr = (long)NN * KI;
    const float* Bm = W + (long)n * KI * Ocols;
    int ntiles = Ocols >> 4;
    for (int t = wave; t < 2 * ntiles; t += 8) {
        int tm = (t / ntiles) * 16;   // batch tile (0 or 16)
        int tn = (t % ntiles) * 16;   // output tile
        v8f acc = {};
        for (int k = 0; k < KI; k += 4) {
            v2f a, bb;
            a.x = A[(long)(tm + r) * Astr + k + 2 * half];
            a.y = A[(long)(tm + r) * Astr + k + 2 * half + 1];
            bb.x = Bm[(long)(k + 2 * half) * Ocols + tn + r];
            bb.y = Bm[(long)(k + 2 * half + 1) * Ocols + tn + r];
            acc = __builtin_amdgcn_wmma_f32_16x16x4_f32(false, a, false, bb, (short)0, acc, false, false);
        }
#pragma unroll
        for (int v = 0; v < 8; v++) {
            int b = tm + v + 8 * half, o = tn + r;
            float val = acc[v] + bias[n * Ocols + o];
            if (act == 1) val = 1.f / (1.f + expf(-val));
            else if (act == 2) val = tanhf(val);
            out[(long)b * NN * outN + (long)n * outN + o] = val;
        }
    }
}

// =====================================================================
// 8) GRU combine: h = r*h + (1-r)*hc,  r = zr[..., H:2H]
// =====================================================================
__global__ __launch_bounds__(256) void hnew_kernel(const float* __restrict__ zr,
                                                   const float* __restrict__ hc,
                                                   float* __restrict__ h) {
    long idx = (long)blockIdx.x * 256 + threadIdx.x;    // B*N*H
    long b = idx / (NN * HH);
    int rem = (int)(idx % (NN * HH));
    int n = rem / HH, o = rem % HH;
    float r = zr[b * (long)NN * (2 * HH) + (long)n * (2 * HH) + HH + o];
    h[idx] = r * h[idx] + (1.f - r) * hc[idx];
}

// =====================================================================
// 9) head: out[b,hor,n,0] = hB[b,n,:]·cw[hor,:] + cb[hor]
// =====================================================================
__global__ __launch_bounds__(256) void head_kernel(const float* __restrict__ hB,
                                                   const float* __restrict__ cw,
                                                   const float* __restrict__ cb,
                                                   float* __restrict__ out) {
    long idx = (long)blockIdx.x * 256 + threadIdx.x;    // B*HOR*N
    int n = (int)(idx % NN);
    int hor = (int)((idx / NN) % HOR);
    long b = idx / (NN * HOR);
    float acc = cb[hor];
#pragma unroll
    for (int q = 0; q < HH; q++)
        acc += hB[b * (long)NN * HH + (long)n * HH + q] * cw[hor * HH + q];
    out[idx] = acc;
}

// =====================================================================
// host side
// =====================================================================
static void gru_step(hipStream_t s, const float* x, long xbstr, int xdim, float* h,
                     int Cp, const float* Wg, const float* Bg,
                     const float* Wu, const float* Bu,
                     const float* tilde, const float* sup2,
                     float* xg, float* zr, float* hc) {
    int n3 = 3 * Cp;
    dim3 blk(32, 8);
    dim3 gsup(4, Cp / 16, NB);
    long bstr = (long)NN * n3;
    // gate path
    concat_kernel<<<dim3(NN, NB), 128, 0, s>>>(x, xbstr, xdim, h, nullptr, xg, Cp);
    gemm512_kernel<<<gsup, blk, 0, s>>>(tilde, xg, bstr, n3, xg + Cp,     bstr, n3, 0);
    gemm512_kernel<<<gsup, blk, 0, s>>>(sup2,  xg, bstr, n3, xg + 2 * Cp, bstr, n3, 0);
    node_gemm_kernel<<<NN, blk, 0, s>>>(xg, Wg, Bg, zr, n3, 2 * HH, 1, 2 * HH);
    // candidate path
    concat_kernel<<<dim3(NN, NB), 128, 0, s>>>(x, xbstr, xdim, h, zr, xg, Cp);
    gemm512_kernel<<<gsup, blk, 0, s>>>(tilde, xg, bstr, n3, xg + Cp,     bstr, n3, 0);
    gemm512_kernel<<<gsup, blk, 0, s>>>(sup2,  xg, bstr, n3, xg + 2 * Cp, bstr, n3, 0);
    node_gemm_kernel<<<NN, blk, 0, s>>>(xg, Wu, Bu, hc, n3, HH, 2, HH);
    // h = r*h + (1-r)*hc
    hnew_kernel<<<(NB * NN * HH) / 256, 256, 0, s>>>(zr, hc, h);
}

extern "C" void kernel_launch(void* const* d_in, const int* in_sizes, int n_in,
                              void* d_out, int out_size, void* d_ws, size_t ws_size,
                              hipStream_t stream) {
    const float* src  = (const float*)d_in[0];   // (B,T,N,DIN)
    const float* gn   = (const float*)d_in[2];   // (N,N,2)
    const float* emb  = (const float*)d_in[3];   // (N,ED)
    const float* gw0  = (const float*)d_in[4];
    const float* gb0  = (const float*)d_in[5];
    const float* uw0  = (const float*)d_in[6];
    const float* ub0  = (const float*)d_in[7];
    const float* gw1  = (const float*)d_in[8];
    const float* gb1  = (const float*)d_in[9];
    const float* uw1  = (const float*)d_in[10];
    const float* ub1  = (const float*)d_in[11];
    const float* cw   = (const float*)d_in[12];  // (HOR,1,1,H)
    const float* cb   = (const float*)d_in[13];
    float* out = (float*)d_out;

    // ---- workspace layout (floats) ----
    float* w = (float*)d_ws;
    long off = 0;
    float* adj   = w + off; off += (long)NN * NN;
    float* dd    = w + off; off += NN;
    float* lam   = w + off; off += 16;
    float* tld   = w + off; off += (long)NN * NN;
    float* sup2  = w + off; off += (long)NN * NN;
    float* W0g   = w + off; off += (long)NN * 3 * CP0 * (2 * HH);
    float* B0g   = w + off; off += (long)NN * 2 * HH;
    float* W0u   = w + off; off += (long)NN * 3 * CP0 * HH;
    float* B0u   = w + off; off += (long)NN * HH;
    float* W1g   = w + off; off += (long)NN * 3 * CP1 * (2 * HH);
    float* B1g   = w + off; off += (long)NN * 2 * HH;
    float* W1u   = w + off; off += (long)NN * 3 * CP1 * HH;
    float* B1u   = w + off; off += (long)NN * HH;
    float* hA    = w + off; off += (long)NB * NN * HH;
    float* hB    = w + off; off += (long)NB * NN * HH;
    float* xg    = w + off; off += (long)NB * NN * 3 * CP1;   // max n3
    float* zr    = w + off; off += (long)NB * NN * 2 * HH;
    float* hc    = w + off; off += (long)NB * NN * HH;

    // ---- graph structure ----
    adj_kernel<<<NN, 256, 0, stream>>>(emb, gn, adj, dd);
    lam_kernel<<<1, 512, 0, stream>>>(dd, lam);
    tilde_kernel<<<(NN * NN) / 256, 256, 0, stream>>>(adj, dd, lam, tld);
    // sup2 = 2*tilde@tilde - I
    gemm512_kernel<<<dim3(4, 32, 1), dim3(32, 8), 0, stream>>>(tld, tld, 0, NN, sup2, 0, NN, 1);

    // ---- per-node weights (loop-invariant; computed once per launch) ----
    build_w_kernel<<<NN, 256, 0, stream>>>(emb, gw0, W0g, DIN + HH, CP0, 2 * HH);
    build_b_kernel<<<NN, 256, 0, stream>>>(emb, gb0, B0g, 2 * HH);
    build_w_kernel<<<NN, 256, 0, stream>>>(emb, uw0, W0u, DIN + HH, CP0, HH);
    build_b_kernel<<<NN, 256, 0, stream>>>(emb, ub0, B0u, HH);
    build_w_kernel<<<NN, 256, 0, stream>>>(emb, gw1, W1g, 2 * HH, CP1, 2 * HH);
    build_b_kernel<<<NN, 256, 0, stream>>>(emb, gb1, B1g, 2 * HH);
    build_w_kernel<<<NN, 256, 0, stream>>>(emb, uw1, W1u, 2 * HH, CP1, HH);
    build_b_kernel<<<NN, 256, 0, stream>>>(emb, ub1, B1u, HH);

    // ---- init hidden states ----
    hipMemsetAsync(hA, 0, sizeof(float) * (long)NB * NN * HH, stream);
    hipMemsetAsync(hB, 0, sizeof(float) * (long)NB * NN * HH, stream);

    // ---- interleaved 2-layer scan over T ----
    for (int t = 0; t < TT; t++) {
        // layer 0: x_t = source[:, t] (B,N,1)
        gru_step(stream, src + (long)t * NN * DIN, (long)TT * NN * DIN, DIN, hA,
                 CP0, W0g, B0g, W0u, B0u, tld, sup2, xg, zr, hc);
        // layer 1: x_t = hA (B,N,H)
        gru_step(stream, hA, (long)NN * HH, HH, hB,
                 CP1, W1g, B1g, W1u, B1u, tld, sup2, xg, zr, hc);
    }

    // ---- output head ----
    head_kernel<<<(NB * HOR * NN) / 256, 256, 0, stream>>>(hB, cw, cb, out);
}